// CortexODE_39204461478331
// MI455X (gfx1250) — compile-verified
//
#include <hip/hip_runtime.h>
#include <hip/hip_bf16.h>

typedef __attribute__((ext_vector_type(2))) float v2f;
typedef __attribute__((ext_vector_type(8))) float v8f;

#define LEAKY 0.2f
#define STEPSZ 0.2f

static __device__ __forceinline__ int imin(int a, int b) { return a < b ? a : b; }

// ---------------- trilinear sample with border clamp ----------------
static __device__ __forceinline__ float trilinear(const float* __restrict__ vol,
                                                  int D, int H, int W,
                                                  float cx, float cy, float cz) {
    float hx = (float)(W - 1), hy = (float)(H - 1), hz = (float)(D - 1);
    float px = fminf(fmaxf((cx + 1.0f) * 0.5f * hx, 0.0f), hx);
    float py = fminf(fmaxf((cy + 1.0f) * 0.5f * hy, 0.0f), hy);
    float pz = fminf(fmaxf((cz + 1.0f) * 0.5f * hz, 0.0f), hz);
    float fx = floorf(px), fy = floorf(py), fz = floorf(pz);
    float wx = px - fx, wy = py - fy, wz = pz - fz;
    int x0 = (int)fx, y0 = (int)fy, z0 = (int)fz;
    int x1 = imin(x0 + 1, W - 1), y1 = imin(y0 + 1, H - 1), z1 = imin(z0 + 1, D - 1);
    const float* pz0 = vol + (size_t)z0 * H * W;
    const float* pz1 = vol + (size_t)z1 * H * W;
    float g000 = pz0[y0 * W + x0], g100 = pz0[y0 * W + x1];
    float g010 = pz0[y1 * W + x0], g110 = pz0[y1 * W + x1];
    float g001 = pz1[y0 * W + x0], g101 = pz1[y0 * W + x1];
    float g011 = pz1[y1 * W + x0], g111 = pz1[y1 * W + x1];
    float c00 = g000 * (1.0f - wx) + g100 * wx;
    float c10 = g010 * (1.0f - wx) + g110 * wx;
    float c01 = g001 * (1.0f - wx) + g101 * wx;
    float c11 = g011 * (1.0f - wx) + g111 * wx;
    float c0 = c00 * (1.0f - wy) + c10 * wy;
    float c1 = c01 * (1.0f - wy) + c11 * wy;
    return c0 * (1.0f - wz) + c1 * wz;
}

// ---------------- 2x2x2 average pooling ----------------
__global__ void pool_kernel(const float* __restrict__ src, float* __restrict__ dst,
                            int D, int H, int W) {  // dst dims
    int i = blockIdx.x * blockDim.x + threadIdx.x;
    int n = D * H * W;
    if (i >= n) return;
    int w = i % W;
    int h = (i / W) % H;
    int d = i / (W * H);
    int SW = W * 2, SH = H * 2;
    size_t b0 = ((size_t)(2 * d) * SH + 2 * h) * SW + 2 * w;
    size_t b1 = b0 + (size_t)SH * SW;
    float s = src[b0] + src[b0 + 1] + src[b0 + SW] + src[b0 + SW + 1]
            + src[b1] + src[b1 + 1] + src[b1 + SW] + src[b1 + SW + 1];
    dst[i] = s * 0.125f;
}

// ---------------- weight prep ----------------
// conv_w (128 x 375) -> convwt (384 x 128), zero padded K
__global__ void prep_convwt_kernel(const float* __restrict__ cw, float* __restrict__ dst) {
    int i = blockIdx.x * blockDim.x + threadIdx.x;
    if (i >= 384 * 128) return;
    int k = i >> 7, o = i & 127;
    dst[i] = (k < 375) ? cw[o * 375 + k] : 0.0f;
}

// src (R x C) -> dst (C x R)
__global__ void transpose_kernel(const float* __restrict__ src, float* __restrict__ dst,
                                 int R, int C) {
    int i = blockIdx.x * blockDim.x + threadIdx.x;
    if (i >= R * C) return;
    int r = i / C, c = i - r * C;
    dst[(size_t)c * R + r] = src[i];
}

// ---------------- sample + conv (WMMA, 1 wave = 16 points) ----------------
__global__ void __launch_bounds__(32)
sample_conv_kernel(const float* __restrict__ x,
                   const float* __restrict__ vol0, const float* __restrict__ vol1,
                   const float* __restrict__ vol2,
                   const float* __restrict__ convwt,  // 384 x 128
                   const float* __restrict__ conv_b,
                   float* __restrict__ zl0) {        // M x 128
    __shared__ float At[16 * 385];  // 16 points x 384 (padded stride 385)
    __shared__ float xs[48];
    int lane = threadIdx.x;
    int m0 = blockIdx.x * 16;
    for (int i = lane; i < 48; i += 32) xs[i] = x[(size_t)m0 * 3 + i];
    __syncthreads();

    const float sc_xy = 208.0f / 176.0f;  // 1/RESCALE for x,z components
    for (int idx = lane; idx < 16 * 384; idx += 32) {
        int p = idx / 384;
        int kk = idx - p * 384;
        float val = 0.0f;
        if (kk < 375) {
            int q = kk / 125;
            int r = kk - q * 125;
            int a = r / 25;
            int b = (r - a * 25) / 5;
            int c = r - a * 25 - b * 5;
            const float* vol;
            int D, H, W;
            if (q == 0)      { vol = vol0; D = 176; H = 208; W = 176; }
            else if (q == 1) { vol = vol1; D = 88;  H = 104; W = 88;  }
            else             { vol = vol2; D = 44;  H = 52;  W = 44;  }
            float scale = (2.0f / 208.0f) * (float)(1 << q);
            float cx = (xs[p * 3 + 0] + (float)(a - 2) * scale) * sc_xy;
            float cy = (xs[p * 3 + 1] + (float)(b - 2) * scale);
            float cz = (xs[p * 3 + 2] + (float)(c - 2) * scale) * sc_xy;
            val = trilinear(vol, D, H, W, cx, cy, cz);
        }
        At[p * 385 + kk] = val;
    }
    __syncthreads();

    int hi = lane >> 4, lo = lane & 15;
    v8f acc[8] = {};
    for (int k0 = 0; k0 < 384; k0 += 4) {
        v2f av;
        av.x = At[lo * 385 + k0 + 2 * hi];
        av.y = At[lo * 385 + k0 + 2 * hi + 1];
#pragma unroll
        for (int t = 0; t < 8; ++t) {
            v2f bv;
            bv.x = convwt[(size_t)(k0 + 2 * hi) * 128 + t * 16 + lo];
            bv.y = convwt[(size_t)(k0 + 2 * hi + 1) * 128 + t * 16 + lo];
            acc[t] = __builtin_amdgcn_wmma_f32_16x16x4_f32(false, av, false, bv,
                                                           (short)0, acc[t], false, false);
        }
    }
#pragma unroll
    for (int t = 0; t < 8; ++t) {
        int n = t * 16 + lo;
        float bb = conv_b[n];
#pragma unroll
        for (int v = 0; v < 8; ++v) {
            int mm = m0 + v + 8 * hi;
            zl0[(size_t)mm * 128 + n] = acc[t][v] + bb;
        }
    }
}

// ---------------- generic fp32 WMMA GEMM: C = act(A*Bt + bias) ----------------
// A: M x K (lda = K), Bt: K x N, writes C[m*ldC + n] (caller pre-offsets C for col offset)
__global__ void __launch_bounds__(32)
gemm_wmma_kernel(const float* __restrict__ A, const float* __restrict__ Bt,
                 const float* __restrict__ bias, float* __restrict__ C,
                 int K, int N, int ldC, int act) {
    int lane = threadIdx.x;
    int hi = lane >> 4, lo = lane & 15;
    int m0 = blockIdx.y * 16;
    int col0 = blockIdx.x * 64;
    size_t arow = (size_t)(m0 + lo) * K;
    v8f acc[4] = {};
    for (int k0 = 0; k0 < K; k0 += 4) {
        v2f av;
        av.x = A[arow + k0 + 2 * hi];
        av.y = A[arow + k0 + 2 * hi + 1];
#pragma unroll
        for (int t = 0; t < 4; ++t) {
            int n = col0 + t * 16 + lo;
            v2f bv;
            bv.x = Bt[(size_t)(k0 + 2 * hi) * N + n];
            bv.y = Bt[(size_t)(k0 + 2 * hi + 1) * N + n];
            acc[t] = __builtin_amdgcn_wmma_f32_16x16x4_f32(false, av, false, bv,
                                                           (short)0, acc[t], false, false);
        }
    }
#pragma unroll
    for (int t = 0; t < 4; ++t) {
        int n = col0 + t * 16 + lo;
        float bb = bias[n];
#pragma unroll
        for (int v = 0; v < 8; ++v) {
            float val = acc[t][v] + bb;
            if (act) val = val > 0.0f ? val : LEAKY * val;
            C[(size_t)(m0 + v + 8 * hi) * ldC + n] = val;
        }
    }
}

// ---------------- fc1: z[:,0:128] = leaky(x @ fc1_w^T + b) ----------------
__global__ void fc1_kernel(const float* __restrict__ x, const float* __restrict__ w,
                           const float* __restrict__ b, float* __restrict__ z, int M) {
    int i = blockIdx.x * blockDim.x + threadIdx.x;
    if (i >= M * 128) return;
    int m = i >> 7, c = i & 127;
    float v = x[m * 3] * w[c * 3] + x[m * 3 + 1] * w[c * 3 + 1]
            + x[m * 3 + 2] * w[c * 3 + 2] + b[c];
    z[(size_t)m * 256 + c] = v > 0.0f ? v : LEAKY * v;
}

// ---------------- fc4 + Euler step: xout = xin + STEP*(h3 @ fc4^T + b) ----------------
__global__ void fc4_step_kernel(const float* __restrict__ h3, const float* __restrict__ w,
                                const float* __restrict__ b, const float* __restrict__ xin,
                                float* __restrict__ xout, int M) {
    int m = blockIdx.x * blockDim.x + threadIdx.x;
    if (m >= M) return;
    const float* h = h3 + (size_t)m * 256;
    float a0 = b[0], a1 = b[1], a2 = b[2];
    for (int k = 0; k < 256; ++k) {
        float hv = h[k];
        a0 += hv * w[k];
        a1 += hv * w[256 + k];
        a2 += hv * w[512 + k];
    }
    xout[m * 3 + 0] = xin[m * 3 + 0] + STEPSZ * a0;
    xout[m * 3 + 1] = xin[m * 3 + 1] + STEPSZ * a1;
    xout[m * 3 + 2] = xin[m * 3 + 2] + STEPSZ * a2;
}

// ---------------- host side ----------------
extern "C" void kernel_launch(void* const* d_in, const int* in_sizes, int n_in,
                              void* d_out, int out_size, void* d_ws, size_t ws_size,
                              hipStream_t stream) {
    const float* x_in   = (const float*)d_in[0];
    const float* V      = (const float*)d_in[1];
    const float* fc1_w  = (const float*)d_in[2];
    const float* fc1_b  = (const float*)d_in[3];
    const float* fc2_w  = (const float*)d_in[4];
    const float* fc2_b  = (const float*)d_in[5];
    const float* fc3_w  = (const float*)d_in[6];
    const float* fc3_b  = (const float*)d_in[7];
    const float* fc4_w  = (const float*)d_in[8];
    const float* fc4_b  = (const float*)d_in[9];
    const float* conv_w = (const float*)d_in[10];
    const float* conv_b = (const float*)d_in[11];
    const float* lfc_w  = (const float*)d_in[12];
    const float* lfc_b  = (const float*)d_in[13];

    const int M = in_sizes[0] / 3;

    const size_t N_VOL1 = 88 * 104 * 88;
    const size_t N_VOL2 = 44 * 52 * 44;
    const size_t N_CWT  = 384 * 128;
    const size_t N_F2T  = 256 * 512;
    const size_t N_F3T  = 512 * 256;
    const size_t N_LFT  = 128 * 128;
    const size_t FIXED  = N_VOL1 + N_VOL2 + N_CWT + N_F2T + N_F3T + N_LFT;
    const size_t PER_PT = 128 + 256 + 512 + 256 + 3 + 3;  // zl0,z,h2,h3,xa,xb

    size_t wsFloats = ws_size / sizeof(float);
    long long cap = ((long long)wsFloats - (long long)FIXED) / (long long)PER_PT;
    if (cap < 16) cap = 16;
    int MC = (int)((cap < (long long)M) ? cap : (long long)M);
    MC &= ~15;
    if (MC < 16) MC = 16;

    float* ws     = (float*)d_ws;
    float* vol1   = ws;
    float* vol2   = vol1 + N_VOL1;
    float* convwt = vol2 + N_VOL2;
    float* fc2t   = convwt + N_CWT;
    float* fc3t   = fc2t + N_F2T;
    float* lfct   = fc3t + N_F3T;
    float* zl0    = lfct + N_LFT;
    float* zbuf   = zl0 + (size_t)MC * 128;
    float* h2     = zbuf + (size_t)MC * 256;
    float* h3     = h2 + (size_t)MC * 512;
    float* xa     = h3 + (size_t)MC * 256;
    float* xb     = xa + (size_t)MC * 3;

    // one-time prep (volume pyramid + weight transposes)
    {
        int n1 = 88 * 104 * 88;
        pool_kernel<<<(n1 + 255) / 256, 256, 0, stream>>>(V, vol1, 88, 104, 88);
        int n2 = 44 * 52 * 44;
        pool_kernel<<<(n2 + 255) / 256, 256, 0, stream>>>(vol1, vol2, 44, 52, 44);
        prep_convwt_kernel<<<(384 * 128 + 255) / 256, 256, 0, stream>>>(conv_w, convwt);
        transpose_kernel<<<(512 * 256 + 255) / 256, 256, 0, stream>>>(fc2_w, fc2t, 512, 256);
        transpose_kernel<<<(256 * 512 + 255) / 256, 256, 0, stream>>>(fc3_w, fc3t, 256, 512);
        transpose_kernel<<<(128 * 128 + 255) / 256, 256, 0, stream>>>(lfc_w, lfct, 128, 128);
    }

    for (int off = 0; off < M; off += MC) {
        int cm = (M - off < MC) ? (M - off) : MC;  // always multiple of 16
        hipMemcpyAsync(xa, x_in + (size_t)off * 3, (size_t)cm * 3 * sizeof(float),
                       hipMemcpyDeviceToDevice, stream);
        float* cur = xa;
        float* nxt = xb;
        for (int s = 0; s < 5; ++s) {
            sample_conv_kernel<<<cm / 16, 32, 0, stream>>>(cur, V, vol1, vol2,
                                                           convwt, conv_b, zl0);
            fc1_kernel<<<(cm * 128 + 255) / 256, 256, 0, stream>>>(cur, fc1_w, fc1_b,
                                                                   zbuf, cm);
            // z_local = zl0 @ lfc^T + b  -> columns 128..255 of z
            gemm_wmma_kernel<<<dim3(2, cm / 16), 32, 0, stream>>>(zl0, lfct, lfc_b,
                                                                  zbuf + 128, 128, 128, 256, 0);
            // h2 = leaky(z @ fc2^T + b)
            gemm_wmma_kernel<<<dim3(8, cm / 16), 32, 0, stream>>>(zbuf, fc2t, fc2_b,
                                                                  h2, 256, 512, 512, 1);
            // h3 = leaky(h2 @ fc3^T + b)
            gemm_wmma_kernel<<<dim3(4, cm / 16), 32, 0, stream>>>(h2, fc3t, fc3_b,
                                                                  h3, 512, 256, 256, 1);
            fc4_step_kernel<<<(cm + 127) / 128, 128, 0, stream>>>(h3, fc4_w, fc4_b,
                                                                  cur, nxt, cm);
            float* t = cur; cur = nxt; nxt = t;
        }
        hipMemcpyAsync((float*)d_out + (size_t)off * 3, cur,
                       (size_t)cm * 3 * sizeof(float), hipMemcpyDeviceToDevice, stream);
    }
}